// ContextAddition_27702539059225
// MI455X (gfx1250) — compile-verified
//
#include <hip/hip_runtime.h>

// Problem constants (from reference)
#define NB      2048
#define SEQ     77
#define DIM     768
#define VOCABSZ 49408
#define EOT_TOK 49407
#define CHUNK   16
#define NCHUNK  5            // ceil(77/16): chunks of 16,16,16,16,13
#define ROWBYTES (DIM * 4)

typedef unsigned int v4u __attribute__((ext_vector_type(4)));
typedef int          v8i __attribute__((ext_vector_type(8)));
typedef int          v4i __attribute__((ext_vector_type(4)));

// -------- Kernel A: find first EOT per batch row (one wave32 per row) -------
__global__ void eot_kernel(const int* __restrict__ tok, int* __restrict__ eot) {
    int wave = threadIdx.x >> 5;
    int lane = threadIdx.x & 31;
    int b = blockIdx.x * (blockDim.x >> 5) + wave;
    if (b >= NB) return;
    const int* row = tok + (size_t)b * SEQ;
    int m = 0x7fffffff;
    for (int p = lane; p < SEQ; p += 32)
        if (row[p] == EOT_TOK) m = min(m, p);
    // wave32 min-reduce
    for (int off = 16; off > 0; off >>= 1)
        m = min(m, __shfl_xor(m, off, 32));
    if (lane == 0) eot[b] = m;
}

// -------- TDM descriptor helpers (CDNA5 ISA ch.8, async_tensor doc §8) -----
// Gather-mode load: pull `nvalid` rows (width DIM fp32) indexed by 16-bit
// row indices from a [tensor_rows x DIM] fp32 tensor at `gbase` into LDS.
__device__ __forceinline__ void tdm_gather_load(const void* gbase,
                                                unsigned tensor_rows,
                                                int nvalid,
                                                unsigned lds_addr,
                                                const unsigned idxw[8]) {
    unsigned long long ga = (unsigned long long)gbase;
    v4u g0;
    g0.x = 1u | (1u << 31);                 // count=1, gather_mode=1, 16b idx
    g0.y = lds_addr;                        // LDS byte address
    g0.z = (unsigned)ga;                    // global_addr[31:0]
    g0.w = (unsigned)(ga >> 32) | (2u << 30); // global_addr[56:32] | type=2
    v8i g1;
    g1[0] = (int)(2u << 16);                // data_size=2 (4 bytes)
    g1[1] = (int)((DIM & 0xFFFFu) << 16);   // tensor_dim0[15:0] @ bits 63:48
    g1[2] = (int)((DIM >> 16) | ((tensor_rows & 0xFFFFu) << 16)); // d0 hi | d1 lo
    g1[3] = (int)((tensor_rows >> 16) | ((unsigned)DIM << 16));   // d1 hi | tile_dim0
    g1[4] = nvalid;                         // tile_dim1 = #valid indices
    g1[5] = DIM;                            // tensor_dim0_stride[31:0] (elements)
    g1[6] = 0;                              // stride hi | dim1_stride lo (unused)
    g1[7] = 0;
    v4i g2; g2[0] = (int)idxw[0]; g2[1] = (int)idxw[1]; g2[2] = (int)idxw[2]; g2[3] = (int)idxw[3];
    v4i g3; g3[0] = (int)idxw[4]; g3[1] = (int)idxw[5]; g3[2] = (int)idxw[6]; g3[3] = (int)idxw[7];
    v8i gx;                                  // trailing/unused operand group
#pragma unroll
    for (int q = 0; q < 8; ++q) gx[q] = 0;
    __builtin_amdgcn_tensor_load_to_lds(g0, g1, g2, g3, gx, 0);
}

// Normal-mode 2D store: write an [nrows x DIM] fp32 tile from LDS to `gdst`
// (contiguous rows, stride DIM elements).
__device__ __forceinline__ void tdm_store_tile(void* gdst, int nrows,
                                               unsigned lds_addr) {
    unsigned long long ga = (unsigned long long)gdst;
    v4u g0;
    g0.x = 1u;                              // count=1, normal mode
    g0.y = lds_addr;
    g0.z = (unsigned)ga;
    g0.w = (unsigned)(ga >> 32) | (2u << 30);
    v8i g1;
    g1[0] = (int)(2u << 16);                // data_size = 4 bytes
    g1[1] = (int)((DIM & 0xFFFFu) << 16);   // tensor_dim0 = DIM
    g1[2] = (int)((DIM >> 16) | (((unsigned)nrows & 0xFFFFu) << 16)); // tensor_dim1 = nrows
    g1[3] = (int)(((unsigned)nrows >> 16) | ((unsigned)DIM << 16));   // tile_dim0 = DIM
    g1[4] = nrows;                          // tile_dim1
    g1[5] = DIM;                            // tensor_dim0_stride
    g1[6] = 0;
    g1[7] = 0;
    v4i z; z[0] = 0; z[1] = 0; z[2] = 0; z[3] = 0;
    v8i gx;
#pragma unroll
    for (int q = 0; q < 8; ++q) gx[q] = 0;
    __builtin_amdgcn_tensor_store_from_lds(g0, g1, z, z, gx, 0);
}

// -------- Kernel B: TDM gather + splice + TDM store, one wave per chunk ----
__global__ __launch_bounds__(32) void gather_kernel(
    const int*   __restrict__ tok,
    const int*   __restrict__ dynb,
    const float* __restrict__ emb,
    const float* __restrict__ ca,
    const float* __restrict__ da,
    const int*   __restrict__ eot,
    float*       __restrict__ out) {
    __shared__ float tile[CHUNK * DIM];     // 48 KB LDS staging tile

    int wid = blockIdx.x;
    int b   = wid / NCHUNK;
    int c   = wid - b * NCHUNK;
    int p0  = c * CHUNK;
    int n   = min(SEQ - p0, CHUNK);

    int e  = __builtin_amdgcn_readfirstlane(eot[b]);
    int dy = __builtin_amdgcn_readfirstlane(dynb[b]);
    int shift = dy ? 16 : 8;
    int e2    = e + 8;                      // start of ca window (if dyn)
    int eend  = e + shift;                  // end of inserted region

    // LDS byte offset of the tile (flat LDS addresses carry offset in [31:0])
    unsigned lds0 = (unsigned)(unsigned long long)(const void*)tile;

    // Build 16x u16 gather indices for the embedding-sourced positions.
    // Positions inside the da/ca window get dummy index 0 and are
    // overwritten in LDS by the follow-up small gathers below.
    const int* trow = tok + (size_t)b * SEQ;
    unsigned idxw[8];
#pragma unroll
    for (int w = 0; w < 8; ++w) idxw[w] = 0u;
    for (int j = 0; j < n; ++j) {
        int p = p0 + j;
        int t = 0;
        if (p < e) {
            t = __builtin_amdgcn_readfirstlane(trow[p]);
        } else if (p >= eend) {
            int sp = p - shift;
            if (sp > SEQ - 1) sp = SEQ - 1;
            t = __builtin_amdgcn_readfirstlane(trow[sp]);
        }
        idxw[j >> 1] |= ((unsigned)t & 0xFFFFu) << ((j & 1) * 16);
    }

    // 1) DMA-gather embedding rows for all n positions of this chunk.
    tdm_gather_load(emb, (unsigned)VOCABSZ, n, lds0, idxw);

    // 2) Overwrite the da sub-run [max(p0,e), min(p0+n, e+8)) in LDS.
    {
        int lo = max(p0, e), hi = min(p0 + n, e2);
        if (lo < hi) {
            unsigned w[8];
#pragma unroll
            for (int q = 0; q < 8; ++q) w[q] = 0u;
            for (int j = 0; j < hi - lo; ++j)
                w[j >> 1] |= ((unsigned)(lo - e + j) & 0xFFFFu) << ((j & 1) * 16);
            tdm_gather_load(da, 8u, hi - lo, lds0 + (unsigned)(lo - p0) * ROWBYTES, w);
        }
    }
    // 3) Overwrite the ca sub-run [max(p0,e+8), min(p0+n, e+16)) if dynamic.
    if (dy) {
        int lo = max(p0, e2), hi = min(p0 + n, e + 16);
        if (lo < hi) {
            unsigned w[8];
#pragma unroll
            for (int q = 0; q < 8; ++q) w[q] = 0u;
            for (int j = 0; j < hi - lo; ++j)
                w[j >> 1] |= ((unsigned)(lo - e2 + j) & 0xFFFFu) << ((j & 1) * 16);
            tdm_gather_load(ca, 8u, hi - lo, lds0 + (unsigned)(lo - p0) * ROWBYTES, w);
        }
    }

    // All tensor DMAs into LDS must land before the tile store reads LDS.
    __builtin_amdgcn_s_wait_tensorcnt(0);

    // 4) DMA the assembled [n x DIM] tile to the contiguous output slice.
    tdm_store_tile(out + ((size_t)b * SEQ + p0) * DIM, n, lds0);
    __builtin_amdgcn_s_wait_tensorcnt(0);
}

extern "C" void kernel_launch(void* const* d_in, const int* in_sizes, int n_in,
                              void* d_out, int out_size, void* d_ws, size_t ws_size,
                              hipStream_t stream) {
    const int*   tok  = (const int*)d_in[0];   // (2048, 77) int32
    const int*   dynb = (const int*)d_in[1];   // (2048,) bool->int
    const float* emb  = (const float*)d_in[2]; // (49408, 768) f32
    const float* ca   = (const float*)d_in[3]; // (8, 768) f32
    const float* da   = (const float*)d_in[4]; // (8, 768) f32
    float* out = (float*)d_out;                // (2048, 77, 768) f32
    int*   eot = (int*)d_ws;                   // 2048 ints of scratch

    // 8 waves per block -> 8 rows per block
    eot_kernel<<<NB / 8, 256, 0, stream>>>(tok, eot);
    // one wave32 workgroup per (batch row, 16-position chunk)
    gather_kernel<<<NB * NCHUNK, 32, 0, stream>>>(tok, dynb, emb, ca, da, eot, out);
}